// RRNetHorosphereInduced_77670188581087
// MI455X (gfx1250) — compile-verified
//
#include <hip/hip_runtime.h>
#include <math.h>

// CDNA5 / gfx1250, wave32. fp32 matrix math via V_WMMA_F32_16X16X4_F32.

typedef __attribute__((ext_vector_type(2))) float v2f;
typedef __attribute__((ext_vector_type(8))) float v8f;
typedef int b128_t __attribute__((vector_size(16)));   // <4 x i32>, matches builtin sig

#define MAXNORM_F 0.99999f   /* 1 - 1e-5 */
#define EPS_F     1e-10f

// gfx1250 async global->LDS path (ASYNCcnt), if the toolchain exposes it.
#if defined(__has_builtin)
#  if __has_builtin(__builtin_amdgcn_global_load_async_to_lds_b128) && \
      __has_builtin(__builtin_amdgcn_s_wait_asynccnt)
#    define USE_ASYNC_LDS 1
#  endif
#endif
#ifndef USE_ASYNC_LDS
#  define USE_ASYNC_LDS 0
#endif

#if USE_ASYNC_LDS
// Issue one async 16B global->LDS copy (tracked on ASYNCcnt).
// Builtin wants (b128_t AS1*, b128_t AS3*, imm offset, imm cpol).
__device__ __forceinline__ void async_copy_b128(const float* gsrc, float* ldst)
{
    __attribute__((address_space(1))) b128_t* g =
        (__attribute__((address_space(1))) b128_t*)
        (const __attribute__((address_space(1))) void*)gsrc;
    __attribute__((address_space(3))) b128_t* l =
        (__attribute__((address_space(3))) b128_t*)
        (__attribute__((address_space(3))) void*)ldst;
    __builtin_amdgcn_global_load_async_to_lds_b128(g, l, 0, 0);
}
#endif

enum { EPI_BIAS = 0, EPI_RELU = 1, EPI_HORO = 2 };

// ---------------------------------------------------------------------------
// Tiled fp32 WMMA GEMM:  C[M,N] = epi( A[M,Kd] * B[Kd,N] )
// 256 threads = 8 wave32; 128x128 tile per workgroup; wave -> 32x64 subtile
// (2x4 grid of 16x16 WMMA accumulators). K staged through double-buffered LDS
// in slabs of 16. Both A and B tiles are stored K-contiguous (B transposed at
// staging time) so every fragment is an aligned 8-byte LDS load -> no VGPR
// repacking in front of the WMMAs.
// Requires M%128==0, N%128==0, Kd%16==0 (true for all shapes here).
// ---------------------------------------------------------------------------
template <int EPI>
__global__ __launch_bounds__(256)
void gemm_wmma_f32(const float* __restrict__ A, const float* __restrict__ B,
                   const float* __restrict__ bias,  // [N]  (HORO: horosphere offsets b)
                   const float* __restrict__ x2,    // [M]  (HORO only: row ||x||^2)
                   float* __restrict__ C,
                   int M, int Kd, int N)
{
    // stride 20 floats: even (keeps b64 alignment) and conflict-free reads
    __shared__ float As[2][128][20];   // A tile, [row][k]
    __shared__ float Bs[2][128][20];   // B tile transposed, [col][k]

    const int t    = threadIdx.x;
    const int wave = t >> 5;
    const int lane = t & 31;
    const int half = lane >> 4;     // 0: lanes 0-15, 1: lanes 16-31
    const int lid  = lane & 15;

    const int m0 = blockIdx.y * 128;
    const int n0 = blockIdx.x * 128;
    const int wm = (wave >> 1) * 32;  // wave row base within tile (0/32/64/96)
    const int wn = (wave & 1) * 64;   // wave col base within tile (0/64)

    v8f acc[2][4] = {};

    // ---- staging helpers (two float4 per thread per tile) ----
    auto stageB_regs = [&](int k0, float4* pb) {
        #pragma unroll
        for (int i = 0; i < 2; ++i) {
            int l = (t + i * 256) * 4;
            int r = l >> 7, c = l & 127;                 // B[k0+r][n0+c..c+3]
            pb[i] = *(const float4*)(B + (size_t)(k0 + r) * N + n0 + c);
        }
    };
    auto storeB_lds = [&](int bsel, const float4* pb) {  // transpose into [col][k]
        #pragma unroll
        for (int i = 0; i < 2; ++i) {
            int l = (t + i * 256) * 4;
            int r = l >> 7, c = l & 127;
            Bs[bsel][c + 0][r] = pb[i].x;
            Bs[bsel][c + 1][r] = pb[i].y;
            Bs[bsel][c + 2][r] = pb[i].z;
            Bs[bsel][c + 3][r] = pb[i].w;
        }
    };
#if USE_ASYNC_LDS
    auto asyncA = [&](int k0, int bsel) {
        #pragma unroll
        for (int i = 0; i < 2; ++i) {
            int l = (t + i * 256) * 4;
            int r = l >> 4, c = l & 15;
            async_copy_b128(A + (size_t)(m0 + r) * Kd + k0 + c,
                            &As[bsel][r][c]);           // (r*20+c)%4==0 -> aligned
        }
    };
#else
    float4 pa[2];
    auto stageA_regs = [&](int k0, float4* pav) {
        #pragma unroll
        for (int i = 0; i < 2; ++i) {
            int l = (t + i * 256) * 4;
            int r = l >> 4, c = l & 15;
            pav[i] = *(const float4*)(A + (size_t)(m0 + r) * Kd + k0 + c);
        }
    };
    auto storeA_lds = [&](int bsel, const float4* pav) {
        #pragma unroll
        for (int i = 0; i < 2; ++i) {
            int l = (t + i * 256) * 4;
            int r = l >> 4, c = l & 15;
            *(float4*)(&As[bsel][r][c]) = pav[i];       // (r*20+c)%4==0 -> aligned
        }
    };
#endif

    // ---- prologue: stage slab 0 into buffer 0 ----
    float4 pb[2];
    stageB_regs(0, pb);
#if USE_ASYNC_LDS
    asyncA(0, 0);
#else
    stageA_regs(0, pa);
    storeA_lds(0, pa);
#endif
    storeB_lds(0, pb);
#if USE_ASYNC_LDS
    __builtin_amdgcn_s_wait_asynccnt(0);
#endif
    __syncthreads();

    // ---- ping-pong main loop, one barrier per slab ----
    int p = 0;
    for (int k0 = 0; k0 < Kd; k0 += 16) {
        const bool more = (k0 + 16) < Kd;
        if (more) {                       // prefetch next slab while computing
            stageB_regs(k0 + 16, pb);
#if USE_ASYNC_LDS
            asyncA(k0 + 16, p ^ 1);
#else
            stageA_regs(k0 + 16, pa);
#endif
        }

        // 4 WMMA K-steps of 4 over the 16-wide slab
        #pragma unroll
        for (int kk = 0; kk < 16; kk += 4) {
            v2f af[2], bf[4];
            // A 16x4 frag (ISA 7.12.2): lanes 0-15 K=kk..kk+1, 16-31 K=kk+2..kk+3
            #pragma unroll
            for (int i = 0; i < 2; ++i)
                af[i] = *(const v2f*)&As[p][wm + i * 16 + lid][kk + 2 * half];
            // B 4x16 frag, symmetric (K contiguous per column in LDS)
            #pragma unroll
            for (int j = 0; j < 4; ++j)
                bf[j] = *(const v2f*)&Bs[p][wn + j * 16 + lid][kk + 2 * half];
            #pragma unroll
            for (int i = 0; i < 2; ++i)
                #pragma unroll
                for (int j = 0; j < 4; ++j)
                    acc[i][j] = __builtin_amdgcn_wmma_f32_16x16x4_f32(
                        /*neg_a=*/false, af[i], /*neg_b=*/false, bf[j],
                        /*c_mod=*/(short)0, acc[i][j],
                        /*reuse_a=*/false, /*reuse_b=*/false);
        }

        if (more) {
#if !USE_ASYNC_LDS
            storeA_lds(p ^ 1, pa);
#endif
            storeB_lds(p ^ 1, pb);
#if USE_ASYNC_LDS
            __builtin_amdgcn_s_wait_asynccnt(0);
#endif
        }
        __syncthreads();
        p ^= 1;
    }

    // ---- fused epilogue; C/D layout: VGPR r -> M = r + half*8, N = lid ----
    #pragma unroll
    for (int i = 0; i < 2; ++i) {
        #pragma unroll
        for (int j = 0; j < 4; ++j) {
            int col = n0 + wn + j * 16 + lid;
            #pragma unroll
            for (int r = 0; r < 8; ++r) {
                int   row = m0 + wm + i * 16 + r + half * 8;
                float val = acc[i][j][r];
                if (EPI == EPI_BIAS) {
                    val += bias[col];
                } else if (EPI == EPI_RELU) {
                    val = fmaxf(val + bias[col], 0.0f);
                } else {  // EPI_HORO: Busemann distance to horosphere
                    float xr = x2[row];
                    float sq = xr - 2.0f * val + 1.0f;          // ||x - omega||^2
                    val = logf(fmaxf(sq, EPS_F)) - logf(fmaxf(1.0f - xr, EPS_F))
                          - bias[col];
                }
                C[(size_t)row * N + col] = val;
            }
        }
    }
}

// ---------------------------------------------------------------------------
// Row-wise: project onto open Poincare ball; also emit ||x||^2 per row.
// One workgroup per row.
// ---------------------------------------------------------------------------
__global__ __launch_bounds__(256)
void projx_rows(const float* __restrict__ in, float* __restrict__ x,
                float* __restrict__ x2out, int D)
{
    __shared__ float red[256];
    const int row = blockIdx.x, t = threadIdx.x;
    const float* p = in + (size_t)row * D;
    float s = 0.f;
    for (int j = t; j < D; j += 256) { float v = p[j]; s += v * v; }
    red[t] = s; __syncthreads();
    for (int off = 128; off > 0; off >>= 1) {
        if (t < off) red[t] += red[t + off];
        __syncthreads();
    }
    float n2 = red[0];
    float sc = fminf(1.0f, MAXNORM_F / fmaxf(sqrtf(n2), EPS_F));
    float* xo = x + (size_t)row * D;
    for (int j = t; j < D; j += 256) xo[j] = p[j] * sc;
    if (t == 0) x2out[row] = n2 * sc * sc;
}

// ---------------------------------------------------------------------------
// Normalize rows of w[i] (ideal points omega) and write transposed [D,K]
// so the feats GEMM reads B row-major. One workgroup per K-row.
// ---------------------------------------------------------------------------
__global__ __launch_bounds__(256)
void normalize_wT(const float* __restrict__ w, float* __restrict__ omegaT,
                  int K, int D)
{
    __shared__ float red[256];
    const int k = blockIdx.x, t = threadIdx.x;
    const float* p = w + (size_t)k * D;
    float s = 0.f;
    for (int j = t; j < D; j += 256) { float v = p[j]; s += v * v; }
    red[t] = s; __syncthreads();
    for (int off = 128; off > 0; off >>= 1) {
        if (t < off) red[t] += red[t + off];
        __syncthreads();
    }
    float inv = 1.0f / fmaxf(sqrtf(red[0]), EPS_F);
    for (int j = t; j < D; j += 256) omegaT[(size_t)j * K + k] = p[j] * inv;
}

// ---------------------------------------------------------------------------
// Fused exponential map + mobius add + projx.  One workgroup per row; D=512
// so each of 256 threads owns exactly 2 elements.
// ---------------------------------------------------------------------------
__global__ __launch_bounds__(256)
void expmap_projx(const float* __restrict__ x, const float* __restrict__ v,
                  float* __restrict__ xout, float* __restrict__ x2out, int D)
{
    __shared__ float rx[256], rv[256], rxv[256];
    const int row = blockIdx.x, t = threadIdx.x;
    const float* xp = x + (size_t)row * D;
    const float* vp = v + (size_t)row * D;

    float xl[2], vl[2];
    float sx = 0.f, sv = 0.f, sxv = 0.f;
    int nl = 0;
    for (int j = t; j < D; j += 256) {
        float a = xp[j], b = vp[j];
        xl[nl] = a; vl[nl] = b; ++nl;
        sx += a * a; sv += b * b; sxv += a * b;
    }
    rx[t] = sx; rv[t] = sv; rxv[t] = sxv; __syncthreads();
    for (int off = 128; off > 0; off >>= 1) {
        if (t < off) { rx[t] += rx[t+off]; rv[t] += rv[t+off]; rxv[t] += rxv[t+off]; }
        __syncthreads();
    }
    const float x2 = rx[0], v2 = rv[0], xv = rxv[0];

    const float lam = 2.0f / fmaxf(1.0f - x2, EPS_F);
    const float vn  = sqrtf(v2);
    const float th  = tanhf(0.5f * lam * vn);
    const float ss  = th / fmaxf(vn, EPS_F);       // y = ss * v
    const float xy  = xv * ss;                     // <x, y>
    const float y2  = v2 * ss * ss;                // ||y||^2
    const float c1  = 1.0f + 2.0f * xy + y2;
    const float c2  = (1.0f - x2) * ss;
    const float den = 1.0f + 2.0f * xy + x2 * y2;
    const float idn = 1.0f / fmaxf(den, EPS_F);

    float rl[2]; float sr = 0.f;
    nl = 0;
    for (int j = t; j < D; j += 256) {
        float r = (c1 * xl[nl] + c2 * vl[nl]) * idn;  // mobius_add(x, y)
        rl[nl] = r; sr += r * r; ++nl;
    }
    __syncthreads();              // rx reuse hazard
    rx[t] = sr; __syncthreads();
    for (int off = 128; off > 0; off >>= 1) {
        if (t < off) rx[t] += rx[t + off];
        __syncthreads();
    }
    const float rn2 = rx[0];
    const float sc  = fminf(1.0f, MAXNORM_F / fmaxf(sqrtf(rn2), EPS_F));
    float* xo = xout + (size_t)row * D;
    nl = 0;
    for (int j = t; j < D; j += 256) { xo[j] = rl[nl] * sc; ++nl; }
    if (t == 0) x2out[row] = rn2 * sc * sc;
}

// ---------------------------------------------------------------------------
extern "C" void kernel_launch(void* const* d_in, const int* in_sizes, int n_in,
                              void* d_out, int out_size, void* d_ws, size_t ws_size,
                              hipStream_t stream)
{
    const float* x_in   = (const float*)d_in[0];
    const float* linW   = (const float*)d_in[1];
    const float* linb   = (const float*)d_in[2];
    const float* w_all  = (const float*)d_in[3];
    const float* b_all  = (const float*)d_in[4];
    const float* W1_all = (const float*)d_in[5];
    const float* b1_all = (const float*)d_in[6];
    const float* W2_all = (const float*)d_in[7];
    const float* b2_all = (const float*)d_in[8];

    const int N = 16384, F = 1024, D = 512, K = 512, H = 1024, Bn = 8;

    // workspace layout (~130 MB)
    char*  ws    = (char*)d_ws;
    float* xbuf  = (float*)ws;  ws += (size_t)N * D * sizeof(float);
    float* x2buf = (float*)ws;  ws += (size_t)N * sizeof(float);
    float* t0    = (float*)ws;  ws += (size_t)N * K * sizeof(float); // feats / pre-projx
    float* hbuf  = (float*)ws;  ws += (size_t)N * H * sizeof(float);
    float* omT   = (float*)ws;  ws += (size_t)D * K * sizeof(float);
    float* vbuf  = t0;          // v[N,D] reuses the dead feats buffer

    const dim3 blk(256);

    // 1) Euclidean linear + projx onto Poincare ball
    gemm_wmma_f32<EPI_BIAS><<<dim3(D / 128, N / 128), blk, 0, stream>>>(
        x_in, linW, linb, nullptr, t0, N, F, D);
    projx_rows<<<N, blk, 0, stream>>>(t0, xbuf, x2buf, D);

    // 2) Riemannian ResNet blocks
    for (int i = 0; i < Bn; ++i) {
        const float* wi = w_all  + (size_t)i * K * D;
        const float* bi = b_all  + (size_t)i * K;
        const float* W1 = W1_all + (size_t)i * K * H;
        const float* b1 = b1_all + (size_t)i * H;
        const float* W2 = W2_all + (size_t)i * H * D;
        const float* b2 = b2_all + (size_t)i * D;

        normalize_wT<<<K, blk, 0, stream>>>(wi, omT, K, D);

        // feats = horo_dist(x, w, b)   (GEMM + fused Busemann epilogue)
        gemm_wmma_f32<EPI_HORO><<<dim3(K / 128, N / 128), blk, 0, stream>>>(
            xbuf, omT, bi, x2buf, t0, N, D, K);

        // h = relu(feats @ W1 + b1)
        gemm_wmma_f32<EPI_RELU><<<dim3(H / 128, N / 128), blk, 0, stream>>>(
            t0, W1, b1, nullptr, hbuf, N, K, H);

        // v = h @ W2 + b2   (overwrites feats buffer)
        gemm_wmma_f32<EPI_BIAS><<<dim3(D / 128, N / 128), blk, 0, stream>>>(
            hbuf, W2, b2, nullptr, vbuf, N, H, D);

        // x = projx(expmap(x, v))
        float* dst = (i == Bn - 1) ? (float*)d_out : xbuf;
        expmap_projx<<<N, blk, 0, stream>>>(xbuf, vbuf, dst, x2buf, D);
    }
}